// GMM1D_26216480375431
// MI455X (gfx1250) — compile-verified
//
#include <hip/hip_runtime.h>
#include <math.h>

typedef float v2f __attribute__((ext_vector_type(2)));
typedef float v8f __attribute__((ext_vector_type(8)));

#define LOG2E_F       1.4426950408889634f
#define INV_SQRT_2PI  0.3989422804014327f

// ---------------------------------------------------------------------------
// Prep kernel: one block of 128 threads.
// Computes per-component constants into d_ws as float4 per k:
//   st[k] = ( mu_k,
//             s_k = -0.5 * exp(-lv_k) * log2(e),
//             t_k = log2( softmax(pi_l)_k * INV_SQRT_2PI * exp(-0.5*lv_k) ),
//             0 )
// so that coef_k * N(mz; mu_k, var_k) = exp2( fma(d*d, s_k, t_k) ), d = mz-mu_k
// ---------------------------------------------------------------------------
__global__ void __launch_bounds__(128) gmm_prep(const float* __restrict__ pi_l,
                                                const float* __restrict__ mu,
                                                const float* __restrict__ lv,
                                                float4* __restrict__ st) {
    __shared__ float red[128];
    const int k = threadIdx.x;

    float p = pi_l[k];

    // max reduction (softmax stability)
    red[k] = p;
    __syncthreads();
    for (int s = 64; s > 0; s >>= 1) {
        if (k < s) red[k] = fmaxf(red[k], red[k + s]);
        __syncthreads();
    }
    float mx = red[0];
    __syncthreads();

    float e = __builtin_amdgcn_exp2f((p - mx) * LOG2E_F);

    // sum reduction
    red[k] = e;
    __syncthreads();
    for (int s = 64; s > 0; s >>= 1) {
        if (k < s) red[k] = red[k] + red[k + s];
        __syncthreads();
    }
    float sum = red[0];

    float pik     = e / sum;
    float lvk     = lv[k];
    float inv_var = __builtin_amdgcn_exp2f(-lvk * LOG2E_F);          // exp(-lv)
    float coef    = pik * INV_SQRT_2PI *
                    __builtin_amdgcn_exp2f(-0.5f * lvk * LOG2E_F);   // pi/(sqrt(2pi)*sigma)

    float s_k = -0.5f * inv_var * LOG2E_F;
    float t_k = __builtin_amdgcn_logf(coef);                          // log2(coef), coef > 0

    st[k] = make_float4(mu[k], s_k, t_k, 0.0f);
}

// ---------------------------------------------------------------------------
// Main kernel: 256 threads (8 waves), each wave produces 64 points.
// Per wave: 4 accumulator tiles (16 points each).  K=128 consumed in chunks
// of 4 via V_WMMA_F32_16X16X4_F32 with B = all-ones (row-sum), coef folded
// into the exp2 argument.  A-matrix f32 16x4 layout (per ISA):
//   VGPR0: lanes0-15 K=k0+0, lanes16-31 K=k0+2
//   VGPR1: lanes0-15 K=k0+1, lanes16-31 K=k0+3
// i.e. lane in half h computes K = 4*chunk + 2h and 4*chunk + 2h + 1.
// ---------------------------------------------------------------------------
__global__ void __launch_bounds__(256) gmm_main(const float* __restrict__ mz,
                                                const float4* __restrict__ st,
                                                float* __restrict__ out,
                                                int n) {
    __shared__ float4 lds_st[128];
    const int tid = threadIdx.x;
    if (tid < 128) lds_st[tid] = st[tid];   // wave-uniform branch in wave32
    __syncthreads();

    const int wave  = tid >> 5;
    const int lane  = tid & 31;
    const int m_idx = lane & 15;
    const int half  = lane >> 4;

    const int base = blockIdx.x * 512 + wave * 64;

    // load the 4 tile points for this lane (lanes 16-31 duplicate 0-15)
    int i0 = base +  0 + m_idx; i0 = i0 < n ? i0 : n - 1;
    int i1 = base + 16 + m_idx; i1 = i1 < n ? i1 : n - 1;
    int i2 = base + 32 + m_idx; i2 = i2 < n ? i2 : n - 1;
    int i3 = base + 48 + m_idx; i3 = i3 < n ? i3 : n - 1;
    const float mz0 = mz[i0];
    const float mz1 = mz[i1];
    const float mz2 = mz[i2];
    const float mz3 = mz[i3];

    v8f c0 = {}; v8f c1 = {}; v8f c2 = {}; v8f c3 = {};
    const v2f bones = {1.0f, 1.0f};

#pragma unroll 4
    for (int chunk = 0; chunk < 32; ++chunk) {
        const int k0 = chunk * 4 + half * 2;
        const float4 q0 = lds_st[k0];       // (mu, s, t, -) for K = k0
        const float4 q1 = lds_st[k0 + 1];   // (mu, s, t, -) for K = k0+1

        v2f a;

        {   // tile 0
            float d0 = mz0 - q0.x, d1 = mz0 - q1.x;
            a.x = __builtin_amdgcn_exp2f(__builtin_fmaf(d0 * d0, q0.y, q0.z));
            a.y = __builtin_amdgcn_exp2f(__builtin_fmaf(d1 * d1, q1.y, q1.z));
            c0 = __builtin_amdgcn_wmma_f32_16x16x4_f32(false, a, false, bones,
                                                       (short)0, c0, false, false);
        }
        {   // tile 1
            float d0 = mz1 - q0.x, d1 = mz1 - q1.x;
            a.x = __builtin_amdgcn_exp2f(__builtin_fmaf(d0 * d0, q0.y, q0.z));
            a.y = __builtin_amdgcn_exp2f(__builtin_fmaf(d1 * d1, q1.y, q1.z));
            c1 = __builtin_amdgcn_wmma_f32_16x16x4_f32(false, a, false, bones,
                                                       (short)0, c1, false, false);
        }
        {   // tile 2
            float d0 = mz2 - q0.x, d1 = mz2 - q1.x;
            a.x = __builtin_amdgcn_exp2f(__builtin_fmaf(d0 * d0, q0.y, q0.z));
            a.y = __builtin_amdgcn_exp2f(__builtin_fmaf(d1 * d1, q1.y, q1.z));
            c2 = __builtin_amdgcn_wmma_f32_16x16x4_f32(false, a, false, bones,
                                                       (short)0, c2, false, false);
        }
        {   // tile 3
            float d0 = mz3 - q0.x, d1 = mz3 - q1.x;
            a.x = __builtin_amdgcn_exp2f(__builtin_fmaf(d0 * d0, q0.y, q0.z));
            a.y = __builtin_amdgcn_exp2f(__builtin_fmaf(d1 * d1, q1.y, q1.z));
            c3 = __builtin_amdgcn_wmma_f32_16x16x4_f32(false, a, false, bones,
                                                       (short)0, c3, false, false);
        }
    }

    // Every column of each D tile is identical (B = ones), so lanes with
    // m_idx==0 hold the results: lane 0 has rows 0-7 (VGPR r = M=r), lane 16
    // has rows 8-15.  Emit two b128 stores per tile from those two lanes.
    if (m_idx == 0) {
        const int rbase = base + half * 8;
        if (rbase + 7 < n) {
            float4* o0 = (float4*)(out + rbase);
            o0[0] = make_float4(c0[0], c0[1], c0[2], c0[3]);
            o0[1] = make_float4(c0[4], c0[5], c0[6], c0[7]);
        }
        if (rbase + 16 + 7 < n) {
            float4* o1 = (float4*)(out + rbase + 16);
            o1[0] = make_float4(c1[0], c1[1], c1[2], c1[3]);
            o1[1] = make_float4(c1[4], c1[5], c1[6], c1[7]);
        }
        if (rbase + 32 + 7 < n) {
            float4* o2 = (float4*)(out + rbase + 32);
            o2[0] = make_float4(c2[0], c2[1], c2[2], c2[3]);
            o2[1] = make_float4(c2[4], c2[5], c2[6], c2[7]);
        }
        if (rbase + 48 + 7 < n) {
            float4* o3 = (float4*)(out + rbase + 48);
            o3[0] = make_float4(c3[0], c3[1], c3[2], c3[3]);
            o3[1] = make_float4(c3[4], c3[5], c3[6], c3[7]);
        }
    }
}

// ---------------------------------------------------------------------------
extern "C" void kernel_launch(void* const* d_in, const int* in_sizes, int n_in,
                              void* d_out, int out_size, void* d_ws, size_t ws_size,
                              hipStream_t stream) {
    const float* mz   = (const float*)d_in[0];
    const float* pi_l = (const float*)d_in[1];
    const float* mu   = (const float*)d_in[2];
    const float* lv   = (const float*)d_in[3];
    float*       out  = (float*)d_out;
    float4*      st   = (float4*)d_ws;   // 128 * 16B = 2 KB of scratch

    gmm_prep<<<1, 128, 0, stream>>>(pi_l, mu, lv, st);

    const int n      = in_sizes[0];            // 4,194,304
    const int blocks = (n + 511) / 512;        // 512 points per block (8 waves x 64)
    gmm_main<<<blocks, 256, 0, stream>>>(mz, st, out, n);
}